// RegionProposalNetwork_17154099380313
// MI455X (gfx1250) — compile-verified
//
#include <hip/hip_runtime.h>
#include <stdint.h>

// ---------------- problem constants ----------------
#define NB     8
#define NANCH  460800
#define PRE    2000
#define PREP   2048          // padded to 32-bit word multiple
#define POSTN  1000
#define CANDC  4096          // candidate capacity per image
#define BINS   4096          // 12-bit key histogram
#define WORDS  64            // PREP/32 suppression-mask words per row

__device__ __forceinline__ float dev_min(float a, float b){ return fminf(a,b); }
__device__ __forceinline__ float dev_max(float a, float b){ return fmaxf(a,b); }

// monotonic key: descending float order == descending unsigned order
__device__ __forceinline__ unsigned fkey(float f){
    unsigned u = __float_as_uint(f);
    return u ^ ((u >> 31) ? 0xFFFFFFFFu : 0x80000000u);
}

__device__ __forceinline__ unsigned ballot32(bool p){
#if __has_builtin(__builtin_amdgcn_ballot_w32)
    return __builtin_amdgcn_ballot_w32(p);
#else
    return (unsigned)__ballot(p);
#endif
}

// ---------------- CDNA5 feature probes ----------------
typedef __attribute__((ext_vector_type(2))) float v2f;
typedef __attribute__((ext_vector_type(8))) float v8f;

#if __has_builtin(__builtin_amdgcn_wmma_f32_16x16x4_f32)
#define HAVE_WMMA_F32X4 1
#endif

#if __has_builtin(__builtin_amdgcn_global_load_async_to_lds_b128)
#define HAVE_ASYNC_LDS 1
// builtin signature (from hipcc diagnostic): param0 = int4 AS1*, param1 = int4 AS3*, imm, imm
typedef int v4i_raw __attribute__((vector_size(4 * sizeof(int))));
typedef __attribute__((address_space(1))) v4i_raw* gas_v4i_p;
typedef __attribute__((address_space(3))) v4i_raw* las_v4i_p;
#endif

// ============================================================
// K1: per-image 4096-bin histogram of score keys (top 12 bits)
// ============================================================
__global__ void k_hist(const float* __restrict__ obj, unsigned* __restrict__ hist){
    const int img = blockIdx.y;
    __shared__ unsigned h[BINS];
    for (int i = threadIdx.x; i < BINS; i += blockDim.x) h[i] = 0;
    __syncthreads();
    const float* o = obj + (size_t)img * NANCH;
    const int stride = gridDim.x * blockDim.x;
    for (int i = blockIdx.x * blockDim.x + threadIdx.x; i < NANCH; i += stride){
        __builtin_prefetch(o + i + 8 * stride, 0, 0);   // global_prefetch_b8
        atomicAdd(&h[fkey(o[i]) >> 20], 1u);
    }
    __syncthreads();
    unsigned* g = hist + (size_t)img * BINS;
    for (int i = threadIdx.x; i < BINS; i += blockDim.x)
        if (h[i]) atomicAdd(&g[i], h[i]);
}

// ============================================================
// K2: find threshold bin (suffix count from top >= PRE)
// ============================================================
__global__ void k_thresh(const unsigned* __restrict__ hist, unsigned* __restrict__ tbin){
    const int img = blockIdx.x;
    if (threadIdx.x == 0){
        const unsigned* h = hist + (size_t)img * BINS;
        unsigned cum = 0;
        int b = BINS - 1;
        for (; b >= 0; --b){ cum += h[b]; if (cum >= PRE) break; }
        tbin[img] = (b < 0) ? 0u : (unsigned)b;
    }
}

// ============================================================
// K3: compact candidates (key>>20 >= tbin) into per-image buffer
// ============================================================
__global__ void k_compact(const float* __restrict__ obj,
                          const unsigned* __restrict__ tbin,
                          unsigned* __restrict__ counter,
                          unsigned long long* __restrict__ cand){
    const int img = blockIdx.y;
    const unsigned tb = tbin[img];
    const float* o = obj + (size_t)img * NANCH;
    const int stride = gridDim.x * blockDim.x;
    for (int i = blockIdx.x * blockDim.x + threadIdx.x; i < NANCH; i += stride){
        __builtin_prefetch(o + i + 8 * stride, 0, 0);
        unsigned k = fkey(o[i]);
        if ((k >> 20) >= tb){
            unsigned pos = atomicAdd(&counter[img], 1u);
            if (pos < CANDC)
                cand[(size_t)img * CANDC + pos] =
                    ((unsigned long long)k << 32) | (unsigned)(~(unsigned)i);
        }
    }
}

// ============================================================
// K4: per-image bitonic sort (4096 u64 in LDS, descending) +
//     gather/decode/clip top-2000, emit boxes + initial keep mask
// ============================================================
__global__ __launch_bounds__(512)
void k_sortdecode(const unsigned long long* __restrict__ cand,
                  const float* __restrict__ anchors,
                  const float* __restrict__ deltas,
                  float4* __restrict__ boxes,
                  unsigned* __restrict__ keep){
    const int img = blockIdx.x;
    const int nt  = blockDim.x;
    __shared__ unsigned long long s[CANDC];          // 32 KB
    for (int i = threadIdx.x; i < CANDC; i += nt)
        s[i] = cand[(size_t)img * CANDC + i];
    __syncthreads();

    // bitonic sort, descending (ties: smaller original idx first via ~idx)
    for (int k = 2; k <= CANDC; k <<= 1){
        for (int j = k >> 1; j > 0; j >>= 1){
            for (int t = threadIdx.x; t < CANDC; t += nt){
                int p = t ^ j;
                if (p > t){
                    bool up = ((t & k) == 0);
                    unsigned long long a = s[t], b = s[p];
                    bool sw = up ? (a < b) : (a > b);
                    if (sw){ s[t] = b; s[p] = a; }
                }
            }
            __syncthreads();
        }
    }

    // decode only the survivors
    const float4* anc4 = (const float4*)anchors;
    const float4* del4 = (const float4*)deltas + (size_t)img * NANCH;
    float4* bx = boxes + (size_t)img * PREP;
    const unsigned lane = threadIdx.x & 31u;
    const float BXC  = 4.135166556742356f;           // log(1000/16)
    const float IMGW = 1024.0f, IMGH = 1024.0f, MINSZ = 1e-3f;

    for (int e = threadIdx.x; e < PREP; e += nt){
        unsigned long long pk = s[e];
        bool valid = (e < PRE) && (pk != 0ull);
        float4 ob = make_float4(0.f, 0.f, 0.f, 0.f);
        bool small = true;
        if (valid){
            unsigned idx = ~(unsigned)(pk & 0xFFFFFFFFull);
            float4 a = anc4[idx];
            float4 d = del4[idx];
            float w  = a.z - a.x, h = a.w - a.y;
            float cx = a.x + 0.5f * w, cy = a.y + 0.5f * h;
            float dw = dev_min(d.z, BXC), dh = dev_min(d.w, BXC);
            float pcx = d.x * w + cx, pcy = d.y * h + cy;
            float pw = expf(dw) * w, ph = expf(dh) * h;
            float x1 = dev_min(dev_max(pcx - 0.5f * pw, 0.f), IMGW);
            float y1 = dev_min(dev_max(pcy - 0.5f * ph, 0.f), IMGH);
            float x2 = dev_min(dev_max(pcx + 0.5f * pw, 0.f), IMGW);
            float y2 = dev_min(dev_max(pcy + 0.5f * ph, 0.f), IMGH);
            ob = make_float4(x1, y1, x2, y2);
            small = ((x2 - x1) < MINSZ) || ((y2 - y1) < MINSZ);
        }
        bx[e] = ob;
        unsigned ball = ballot32(valid && !small);
        if (lane == 0) keep[(size_t)img * WORDS + (e >> 5)] = ball;
    }
}

// ============================================================
// K5: suppression-bitmask tiles. One wave32 per 16-row x 32-col
// tile. Pairwise (area_i + area_j) via V_WMMA_F32_16X16X4_F32
// (rank-2 outer product, co-executes with VALU intersection math).
// Box staging via async global->LDS when available.
// ============================================================
__global__ __launch_bounds__(32)
void k_nms_tiles(const float4* __restrict__ boxes, unsigned* __restrict__ mask){
    const int img = blockIdx.z;
    const int r0  = blockIdx.y * 16;          // row tile base
    const int w   = blockIdx.x;               // col word index (32 cols)
    const int c0  = w * 32;
    const unsigned lane = threadIdx.x;        // 0..31
    const float NMS_T = 0.7f;

    __shared__ float4 ldsRow[16];
    __shared__ float4 ldsCol[32];
    const float4* bx = boxes + (size_t)img * PREP;

#if defined(HAVE_ASYNC_LDS)
    __builtin_amdgcn_global_load_async_to_lds_b128(
        (gas_v4i_p)(bx + c0 + lane), (las_v4i_p)&ldsCol[lane], 0, 0);
    if (lane < 16)
        __builtin_amdgcn_global_load_async_to_lds_b128(
            (gas_v4i_p)(bx + r0 + lane), (las_v4i_p)&ldsRow[lane], 0, 0);
    asm volatile("s_wait_asynccnt 0" ::: "memory");
#else
    ldsCol[lane] = bx[c0 + lane];
    if (lane < 16) ldsRow[lane] = bx[r0 + lane];
#endif
    __syncthreads();

    // Build WMMA operands: D[i][j] = area_row[i]*1 + 1*area_col[j]
    v2f a, b0, b1;
    if (lane < 16){
        float4 rb = ldsRow[lane];
        a.x = (rb.z - rb.x) * (rb.w - rb.y);  a.y = 1.0f;      // A: K=0 area, K=1 one
        float4 cA = ldsCol[lane];
        b0.x = 1.0f;  b0.y = (cA.z - cA.x) * (cA.w - cA.y);    // B: K=0 one, K=1 area
        float4 cB = ldsCol[lane + 16];
        b1.x = 1.0f;  b1.y = (cB.z - cB.x) * (cB.w - cB.y);
    } else {
        a.x = 0.f; a.y = 0.f; b0.x = 0.f; b0.y = 0.f; b1.x = 0.f; b1.y = 0.f; // K=2,3 zero
    }

#if defined(HAVE_WMMA_F32X4)
    v8f cz = {};
    v8f sum0 = __builtin_amdgcn_wmma_f32_16x16x4_f32(false, a, false, b0,
                                                     (short)0, cz, false, false);
    v8f sum1 = __builtin_amdgcn_wmma_f32_16x16x4_f32(false, a, false, b1,
                                                     (short)0, cz, false, false);
#endif

    const int  sub = (lane >= 16) ? 8 : 0;    // D layout: lanes16-31 hold rows M+8
    const int  cl  = lane & 15;               // D layout: N = lane (per half)
    const float4 cb0 = ldsCol[cl];
    const float4 cb1 = ldsCol[cl + 16];
    unsigned* mrow = mask + ((size_t)img * PREP + r0) * WORDS + w;

#pragma unroll
    for (int r = 0; r < 8; ++r){
        const int rloc = r + sub;
        float4 rb = ldsRow[rloc];
        float ix0 = dev_min(rb.z, cb0.z) - dev_max(rb.x, cb0.x);
        float iy0 = dev_min(rb.w, cb0.w) - dev_max(rb.y, cb0.y);
        float in0 = dev_max(ix0, 0.f) * dev_max(iy0, 0.f);
        float ix1 = dev_min(rb.z, cb1.z) - dev_max(rb.x, cb1.x);
        float iy1 = dev_min(rb.w, cb1.w) - dev_max(rb.y, cb1.y);
        float in1 = dev_max(ix1, 0.f) * dev_max(iy1, 0.f);
#if defined(HAVE_WMMA_F32X4)
        float s0 = sum0[r], s1 = sum1[r];
#else
        float ar = (rb.z - rb.x) * (rb.w - rb.y);
        float s0 = ar + (cb0.z - cb0.x) * (cb0.w - cb0.y);
        float s1 = ar + (cb1.z - cb1.x) * (cb1.w - cb1.y);
#endif
        // iou > T  <=>  inter*(1+T) > T*(a_i + a_j)   (division-free)
        bool p0 = in0 * (1.0f + NMS_T) > NMS_T * s0;
        bool p1 = in1 * (1.0f + NMS_T) > NMS_T * s1;
        unsigned m0 = ballot32(p0);
        unsigned m1 = ballot32(p1);
        unsigned lowW  = (m0 & 0xFFFFu) | ((m1 & 0xFFFFu) << 16);   // row r0+r
        unsigned highW = (m0 >> 16)     | (m1 & 0xFFFF0000u);       // row r0+r+8
        if (lane == 0)  mrow[(size_t)r * WORDS]       = lowW;
        if (lane == 16) mrow[(size_t)(r + 8) * WORDS] = highW;
    }
}

// ============================================================
// K6: serial greedy scan — single wave per image, keep bits in
// registers (lane owns words lane and lane+32), no barriers.
// ============================================================
__global__ __launch_bounds__(32)
void k_nms_scan(const unsigned* __restrict__ maskg, unsigned* __restrict__ keep){
    const int img = blockIdx.x;
    const int lane = threadIdx.x;
    unsigned* kw = keep + (size_t)img * WORDS;
    unsigned kA = kw[lane];
    unsigned kB = kw[lane + 32];
    const unsigned* m = maskg + (size_t)img * PREP * WORDS;
    const int wA = lane, wB = lane + 32;

    for (int i = 0; i < PRE; ++i){
        const int wi = i >> 5, bi = i & 31;
        unsigned wv = (wi < 32) ? (unsigned)__shfl((int)kA, wi, 32)
                                : (unsigned)__shfl((int)kB, wi - 32, 32);
        if ((wv >> bi) & 1u){
            const unsigned* row = m + (size_t)i * WORDS;
            unsigned m0 = row[lane], m1 = row[lane + 32];
            unsigned cutA = (wA < wi) ? 0u : ((wA > wi) ? 0xFFFFFFFFu : (0xFFFFFFFEu << bi));
            unsigned cutB = (wB < wi) ? 0u : ((wB > wi) ? 0xFFFFFFFFu : (0xFFFFFFFEu << bi));
            kA &= ~(m0 & cutA);
            kB &= ~(m1 & cutB);
        }
    }
    kw[lane] = kA;
    kw[lane + 32] = kB;
}

// ============================================================
// K7: emit 1000 boxes: kept entries in score order, then
// suppressed entries in index order (matches top_k stability).
// ============================================================
__global__ __launch_bounds__(32)
void k_output(const float4* __restrict__ boxes, const unsigned* __restrict__ keep,
              float4* __restrict__ out){
    const int img = blockIdx.x;
    const int lane = threadIdx.x;
    const unsigned* kw = keep + (size_t)img * WORDS;
    const int wA = lane, wB = lane + 32;
    auto validMask = [](int w)->unsigned{
        return (w < 62) ? 0xFFFFFFFFu : ((w == 62) ? 0x0000FFFFu : 0u);  // e < 2000
    };
    unsigned kA = kw[wA] & validMask(wA);
    unsigned kB = kw[wB] & validMask(wB);
    unsigned noA = validMask(wA) & ~kA;
    unsigned noB = validMask(wB) & ~kB;

    // wave-level inclusive scans of popcounts
    unsigned pA = __popc(kA), pB = __popc(kB), qA = __popc(noA), qB = __popc(noB);
    unsigned sA = pA, sB = pB, tA = qA, tB = qB;
    for (int o = 1; o < 32; o <<= 1){
        unsigned u;
        u = (unsigned)__shfl_up((int)sA, o, 32); if (lane >= o) sA += u;
        u = (unsigned)__shfl_up((int)sB, o, 32); if (lane >= o) sB += u;
        u = (unsigned)__shfl_up((int)tA, o, 32); if (lane >= o) tA += u;
        u = (unsigned)__shfl_up((int)tB, o, 32); if (lane >= o) tB += u;
    }
    unsigned totA  = (unsigned)__shfl((int)sA, 31, 32);
    unsigned totB  = (unsigned)__shfl((int)sB, 31, 32);
    unsigned totNA = (unsigned)__shfl((int)tA, 31, 32);
    unsigned nKept = totA + totB;
    unsigned exKA = sA - pA;                 // kept before word wA
    unsigned exKB = totA + (sB - pB);
    unsigned exNA = tA - qA;
    unsigned exNB = totNA + (tB - qB);

    const float4* bx = boxes + (size_t)img * PREP;
    float4* o4 = out + (size_t)img * POSTN;

    unsigned cnt, r;
    cnt = exKA; r = kA;
    while (r){ int b = __ffs(r) - 1; r &= r - 1;
        if (cnt < POSTN) o4[cnt] = bx[wA * 32 + b]; ++cnt; }
    cnt = exKB; r = kB;
    while (r){ int b = __ffs(r) - 1; r &= r - 1;
        if (cnt < POSTN) o4[cnt] = bx[wB * 32 + b]; ++cnt; }
    cnt = nKept + exNA; r = noA;
    while (r){ int b = __ffs(r) - 1; r &= r - 1;
        if (cnt < POSTN) o4[cnt] = bx[wA * 32 + b]; ++cnt; }
    cnt = nKept + exNB; r = noB;
    while (r){ int b = __ffs(r) - 1; r &= r - 1;
        if (cnt < POSTN) o4[cnt] = bx[wB * 32 + b]; ++cnt; }
}

// ============================================================
// host launcher
// ============================================================
extern "C" void kernel_launch(void* const* d_in, const int* in_sizes, int n_in,
                              void* d_out, int out_size, void* d_ws, size_t ws_size,
                              hipStream_t stream) {
    (void)in_sizes; (void)n_in; (void)out_size; (void)ws_size;
    const float* anchors = (const float*)d_in[0];   // [N,4]
    const float* obj     = (const float*)d_in[1];   // [B,N]
    const float* deltas  = (const float*)d_in[2];   // [B,N,4]

    auto al = [](size_t x){ return (x + 255) & ~(size_t)255; };
    char* ws = (char*)d_ws;
    size_t off = 0;
    unsigned*           hist    = (unsigned*)(ws + off);           off += al((size_t)NB * BINS * 4);
    unsigned*           counter = (unsigned*)(ws + off);           off += al((size_t)NB * 4);
    unsigned long long* cand    = (unsigned long long*)(ws + off);
    size_t zeroBytes = off + (size_t)NB * CANDC * 8;               off += al((size_t)NB * CANDC * 8);
    unsigned*           tbin    = (unsigned*)(ws + off);           off += al((size_t)NB * 4);
    float4*             boxes   = (float4*)(ws + off);             off += al((size_t)NB * PREP * 16);
    unsigned*           keep    = (unsigned*)(ws + off);           off += al((size_t)NB * WORDS * 4);
    unsigned*           mask    = (unsigned*)(ws + off);           off += al((size_t)NB * PREP * WORDS * 4);
    (void)mask;

    // zero hist + counters + candidate buffer (graph-capturable)
    (void)hipMemsetAsync(d_ws, 0, zeroBytes, stream);

    dim3 gStream(120, NB);
    k_hist   <<<gStream, 256, 0, stream>>>(obj, hist);
    k_thresh <<<NB, 32, 0, stream>>>(hist, tbin);
    k_compact<<<gStream, 256, 0, stream>>>(obj, tbin, counter, cand);
    k_sortdecode<<<NB, 512, 0, stream>>>(cand, anchors, deltas, boxes, keep);
    dim3 gTiles(WORDS, PREP / 16, NB);
    k_nms_tiles<<<gTiles, 32, 0, stream>>>(boxes, mask);
    k_nms_scan <<<NB, 32, 0, stream>>>(mask, keep);
    k_output   <<<NB, 32, 0, stream>>>(boxes, keep, (float4*)d_out);
}